// RawSampleEmbedding_56307021250866
// MI455X (gfx1250) — compile-verified
//
#include <hip/hip_runtime.h>
#include <math.h>

// Problem: x [32,8192] f32, W [256,256] f32, b [256] f32
// out [32,256,8192] f32 == flat [N=262144, D=256] row-major.
// Reference's softmax(-(d^2)*1e12) is a hard one-hot -> column gather of W.
// Roofline: 256 MB of output stores / 23.3 TB/s ~= 11 us; compute trivial.
// WMMA would turn the O(N*D) gather into an O(N*C*D) matmul (256x the math),
// so the CDNA5-specific paths used are async global->LDS staging + LDS gather.
#define D_DIM   256
#define C_DIM   256
#define PITCH   260                    // LDS row pitch (floats): 16B-aligned rows for all c,
                                       // conflict-free b128 gathers (bank = 4c+4l+j mod 64)
#define WT_WORDS (C_DIM * PITCH)       // 66560 floats for transposed W
#define B_OFF    WT_WORDS              // bias region starts here (16B aligned)
#define LDS_WORDS (WT_WORDS + D_DIM)   // 66816 floats = 267264 B (< 320 KB/WGP)
#define BLOCK   1024
#define GRID    256

__global__ __launch_bounds__(BLOCK)
void mulaw_embed_kernel(const float* __restrict__ x,
                        const float* __restrict__ W,
                        const float* __restrict__ b,
                        float* __restrict__ out,
                        int nSamples)
{
    extern __shared__ float sm[];
    const int tau  = threadIdx.x;
    const int lane = tau & 31;
    const int wav  = tau >> 5;

    // ---- Stage bias b[256] into LDS with CDNA5 async global->LDS copy.
    // Threads 0..63 each move 16 B; tracked by ASYNCcnt.
    if (tau < 64) {
        unsigned ldsAddr = (unsigned)(uintptr_t)(sm + B_OFF + tau * 4); // LDS byte offset
        unsigned gOff    = (unsigned)(tau * 16);
        asm volatile("global_load_async_to_lds_b128 %0, %1, %2"
                     :: "v"(ldsAddr), "v"(gOff), "s"(b)
                     : "memory");
    }

    // ---- Stage W transposed into LDS: Wt[c][d] = W[d*256 + c], pitch PITCH.
    // Wave 'wav' handles spans p = wav + 32q (128 contiguous floats each);
    // lane l takes elements l + 32j of the span: global b32 loads stay
    // perfectly coalesced and the transposed ds_store bank index is
    // (4l + d) mod 64 -> only 2-way conflicted.
    {
        #pragma unroll 4
        for (int q = 0; q < 16; ++q) {
            const int p  = wav + 32 * q;     // span index (128 floats per span)
            const int d  = p >> 1;           // W row (constant within span)
            const int cb = (p & 1) * 128;    // W col base of this span
            #pragma unroll
            for (int j = 0; j < 4; ++j) {
                const int c = cb + lane + 32 * j;
                sm[c * PITCH + d] = W[p * 128 + lane + 32 * j];
            }
        }
    }

    // Ensure our wave's async LDS writes landed, then workgroup barrier.
    asm volatile("s_wait_asynccnt 0x0" ::: "memory");
    __syncthreads();

    // ---- Per-thread bias cache: lane handles dims d = 4*lane + 128k + j.
    const float4* __restrict__ B4 = (const float4*)(sm + B_OFF);
    const float4 bk0 = B4[lane];
    const float4 bk1 = B4[32 + lane];

    const float INV_LOG256 = 0.18033688011112042f;  // 1 / ln(256)
    const int numChunks  = nSamples >> 5;           // 32 samples per wave-chunk
    const int totalWaves = gridDim.x * (BLOCK / 32);
    const int gw         = blockIdx.x * (BLOCK / 32) + wav;

    for (int chunk = gw; chunk < numChunks; chunk += totalWaves) {
        const int n0 = chunk << 5;

        // Each lane: hard one-hot class index for sample n0+lane.
        float xv = x[n0 + lane];
        float mu = copysignf(log1pf(255.0f * fabsf(xv)) * INV_LOG256, xv);
        float scaled = (mu + 1.0f) * 128.0f;
        int ci = (int)rintf(scaled);                // argmin_y (scaled - y)^2
        ci = ci < 0 ? 0 : (ci > 255 ? 255 : ci);

        // Whole wave cooperates on one sample at a time.
        #pragma unroll 2
        for (int s = 0; s < 32; ++s) {
            const int c = __shfl(ci, s, 32);        // broadcast class index
            const float4* __restrict__ row = (const float4*)(sm + c * PITCH);

            // Two conflict-free ds_load_b128 gathers of W column c.
            float4 w0 = row[lane];
            float4 w1 = row[32 + lane];

            float z0x = w0.x + bk0.x, z0y = w0.y + bk0.y;
            float z0z = w0.z + bk0.z, z0w = w0.w + bk0.w;
            float z1x = w1.x + bk1.x, z1y = w1.y + bk1.y;
            float z1z = w1.z + bk1.z, z1w = w1.w + bk1.w;

            float ss = z0x * z0x;
            ss = fmaf(z0y, z0y, ss);
            ss = fmaf(z0z, z0z, ss);
            ss = fmaf(z0w, z0w, ss);
            ss = fmaf(z1x, z1x, ss);
            ss = fmaf(z1y, z1y, ss);
            ss = fmaf(z1z, z1z, ss);
            ss = fmaf(z1w, z1w, ss);

            // wave32 cross-lane reduction of sum of squares
            #pragma unroll
            for (int off = 16; off > 0; off >>= 1)
                ss += __shfl_xor(ss, off, 32);

            const float inv = __builtin_amdgcn_rsqf(ss);   // v_rsq_f32

            float4 o0, o1;
            o0.x = z0x * inv; o0.y = z0y * inv; o0.z = z0z * inv; o0.w = z0w * inv;
            o1.x = z1x * inv; o1.y = z1y * inv; o1.z = z1z * inv; o1.w = z1w * inv;

            // Two fully coalesced 512B global_store_b128 per sample.
            float4* __restrict__ o = (float4*)(out + (size_t)(n0 + s) * D_DIM);
            o[lane]      = o0;
            o[32 + lane] = o1;
        }
    }
}

extern "C" void kernel_launch(void* const* d_in, const int* in_sizes, int n_in,
                              void* d_out, int out_size, void* d_ws, size_t ws_size,
                              hipStream_t stream) {
    const float* x = (const float*)d_in[0];   // [32, 8192]
    const float* W = (const float*)d_in[1];   // [256, 256]
    const float* b = (const float*)d_in[2];   // [256]
    float* out = (float*)d_out;               // [32, 256, 8192] flat

    const int nSamples = in_sizes[0];         // 262144
    const size_t ldsBytes = (size_t)LDS_WORDS * sizeof(float); // 267264 B

    hipLaunchKernelGGL(mulaw_embed_kernel, dim3(GRID), dim3(BLOCK), ldsBytes, stream,
                       x, W, b, out, nSamples);
}